// LightweightBEVMambaBlock_58841051955477
// MI455X (gfx1250) — compile-verified
//
#include <hip/hip_runtime.h>
#include <hip/hip_bf16.h>

// ---------------------------------------------------------------------------
// LightweightBEVMambaBlock for MI455X (gfx1250, wave32).
// stats -> norm+pool -> w_in WMMA GEMM -> dwconv+silu -> row proj GEMMs
// -> row scans -> col proj GEMMs -> col scans -> gated w_out WMMA GEMM
// -> bilinear x2 upsample + residual.
// GEMMs: V_WMMA_F32_16X16X4_F32, 16x64 tile per wave (4 accumulators),
// weights staged to LDS via Tensor Data Mover (TDM, 6-arg builtin).
// ---------------------------------------------------------------------------

typedef __attribute__((ext_vector_type(2))) float v2f;
typedef __attribute__((ext_vector_type(8))) float v8f;
typedef __attribute__((ext_vector_type(4))) float v4fv;
typedef __attribute__((ext_vector_type(4))) unsigned int v4u;
typedef __attribute__((ext_vector_type(8))) int v8i;
typedef __attribute__((ext_vector_type(4))) int v4i;

#ifndef __has_builtin
#define __has_builtin(x) 0
#endif
#if __has_builtin(__builtin_amdgcn_wmma_f32_16x16x4_f32)
#define USE_WMMA_F32X4 1
#else
#define USE_WMMA_F32X4 0
#endif
#if __has_builtin(__builtin_amdgcn_tensor_load_to_lds) && __has_builtin(__builtin_amdgcn_s_wait_tensorcnt)
#define USE_TDM 1
#else
#define USE_TDM 0
#endif

#define NB   4
#define CDIM 256
#define HH   200
#define WW   200
#define HS   100
#define WS   100
#define INNER 128
#define NPOS (NB * HS * WS)           // 40000
#define NPER ((size_t)CDIM * HH * WW) // 10,240,000 per batch

__device__ __forceinline__ float sigf(float v) {
    // fast sigmoid: v_exp_f32 + v_rcp_f32 (HBM-bound op; fast-math is fine)
    return __builtin_amdgcn_rcpf(1.0f + __expf(-v));
}

// ---------------------------------------------------------------- stats -----
__global__ void zero_stats_kernel(float* stats) {
    if (threadIdx.x < 16) stats[threadIdx.x] = 0.0f;
}

__global__ void reduce_stats_kernel(const float* __restrict__ x, float* __restrict__ stats) {
    const int BLOCKS_PER_BATCH = 160;
    int batch = blockIdx.x / BLOCKS_PER_BATCH;
    int chunk = blockIdx.x % BLOCKS_PER_BATCH;
    size_t per_block = NPER / BLOCKS_PER_BATCH; // 64000
    size_t start = (size_t)batch * NPER + (size_t)chunk * per_block;
    float s = 0.0f, ss = 0.0f;
    for (size_t i = threadIdx.x; i < per_block; i += blockDim.x) {
        float v = x[start + i];
        s += v; ss += v * v;
    }
    __shared__ float ls[256], lss[256];
    ls[threadIdx.x] = s; lss[threadIdx.x] = ss;
    __syncthreads();
    for (int off = 128; off > 0; off >>= 1) {
        if ((int)threadIdx.x < off) {
            ls[threadIdx.x]  += ls[threadIdx.x + off];
            lss[threadIdx.x] += lss[threadIdx.x + off];
        }
        __syncthreads();
    }
    if (threadIdx.x == 0) {
        atomicAdd(&stats[batch],     ls[0]);
        atomicAdd(&stats[4 + batch], lss[0]);
    }
}

// ------------------------------------------------------- norm + 2x2 pool ----
__global__ void pool_norm_kernel(const float* __restrict__ x, const float* __restrict__ stats,
                                 const float* __restrict__ gnw, const float* __restrict__ gnb,
                                 float* __restrict__ pooled) {
    size_t idx = (size_t)blockIdx.x * blockDim.x + threadIdx.x;
    if (idx >= (size_t)NB * CDIM * HS * WS) return;
    int j = (int)(idx % WS);  size_t t = idx / WS;
    int i = (int)(t % HS);    t /= HS;
    int c = (int)(t % CDIM);  int b = (int)(t / CDIM);

    const float inv = 1.0f / (float)NPER;
    float mean = stats[b] * inv;
    float var  = stats[4 + b] * inv - mean * mean;
    float rstd = rsqrtf(var + 1e-5f);

    const float* xp = x + ((size_t)b * CDIM + c) * (HH * WW);
    int yy = 2 * i, xx = 2 * j;
    float a = xp[(size_t)yy * WW + xx]       + xp[(size_t)yy * WW + xx + 1] +
              xp[(size_t)(yy + 1) * WW + xx] + xp[(size_t)(yy + 1) * WW + xx + 1];
    a *= 0.25f;
    a = (a - mean) * rstd * gnw[c] + gnb[c];
    size_t p = ((size_t)b * HS + i) * WS + j;
    pooled[p * CDIM + c] = a;
}

// ---------------------------------------------------------- WMMA GEMM -------
// D[p, n] = ACT( sum_k A[p,k] * W[n,k] + bias[n] ), optional gate on A.
// Block = 128 threads (4 waves). Each wave: 16(M) x 64(N), 4 f32 accumulators,
// K stepped by 4 via V_WMMA_F32_16X16X4_F32. Weight tile 64 x K staged in LDS
// by the Tensor Data Mover (contiguous 1-D descriptor).
template <int KT, bool GATE, int ACT, bool PLANAR>
__global__ void gemm_wmma_kernel(const float* __restrict__ A,
                                 const float* __restrict__ W,
                                 const float* __restrict__ bias,
                                 const float* __restrict__ gate, // stride CDIM, pre-offset
                                 float* __restrict__ D, int N) {
    __shared__ float lds_w[64 * KT];
    const int tid   = threadIdx.x;
    const int lane  = tid & 31;
    const int wv    = tid >> 5;                 // 0..3
    const int nbase = blockIdx.y * 64;
    const int p0    = (blockIdx.x * 4 + wv) * 16;

#if USE_TDM
    if (wv == 0) {
        const unsigned ntot = 64u * (unsigned)KT;                    // elements (<= 16384)
        unsigned long long ga = (unsigned long long)(const void*)(W + (size_t)nbase * KT);
        // generic shared address: low 32 bits == LDS byte offset (ISA flat-aperture rule)
        unsigned lds_off = (unsigned)(unsigned long long)(const void*)&lds_w[0];
        v4u g0;
        g0.x = 1u;                                                   // count=1 (valid user D#)
        g0.y = lds_off;                                              // lds_addr
        g0.z = (unsigned)(ga & 0xFFFFFFFFu);                         // global_addr[31:0]
        g0.w = (unsigned)((ga >> 32) & 0x01FFFFFFu) | (2u << 30);    // addr[56:32] | type=2
        v8i g1;
        g1[0] = (int)(2u << 16);                                     // data_size = 4B
        g1[1] = (int)((ntot & 0xFFFFu) << 16);                       // tensor_dim0[15:0]
        g1[2] = (int)((ntot >> 16) & 0xFFFFu);                       // tensor_dim0[31:16]
        g1[3] = (int)((ntot & 0xFFFFu) << 16);                       // tile_dim0
        g1[4] = 0;                                                   // tile_dim1/2 unused (1-D)
        g1[5] = (int)ntot;                                           // tensor_dim0_stride lo
        g1[6] = 0;
        g1[7] = 0;
        v4i zz4 = {0, 0, 0, 0};
        v8i zz8 = {0, 0, 0, 0, 0, 0, 0, 0};
        __builtin_amdgcn_tensor_load_to_lds(g0, g1, zz4, zz4, zz8, 0);
        __builtin_amdgcn_s_wait_tensorcnt(0);
    }
#else
    {   // cooperative vectorized copy of the contiguous 64 x KT tile
        const v4fv* src = (const v4fv*)(W + (size_t)nbase * KT);
        v4fv* dst = (v4fv*)&lds_w[0];
        for (int idx = tid; idx < (64 * KT) / 4; idx += blockDim.x) dst[idx] = src[idx];
    }
#endif
    __syncthreads();

    const int m    = lane & 15;
    const int half = lane >> 4;                 // 0 or 1
    const size_t rowA = (size_t)(p0 + m) * KT;

    v8f acc[4];
    #pragma unroll
    for (int nn = 0; nn < 4; ++nn)
        acc[nn] = (v8f){0.f, 0.f, 0.f, 0.f, 0.f, 0.f, 0.f, 0.f};

    #pragma unroll 4
    for (int k0 = 0; k0 < KT; k0 += 4) {
        const int ka = k0 + 2 * half;
        v2f a = *(const v2f*)(A + rowA + ka);
        if (GATE) {
            v2f g = *(const v2f*)(gate + (size_t)(p0 + m) * CDIM + ka);
            a.x *= sigf(g.x);
            a.y *= sigf(g.y);
        }
#if USE_WMMA_F32X4
        #pragma unroll
        for (int nn = 0; nn < 4; ++nn) {
            v2f bvec = *(const v2f*)(&lds_w[(nn * 16 + m) * KT + ka]);
            acc[nn] = __builtin_amdgcn_wmma_f32_16x16x4_f32(
                false, a, false, bvec, (short)0, acc[nn], false, false);
        }
#else
        #pragma unroll
        for (int nn = 0; nn < 4; ++nn) {
            // scalar fallback placeholder; real path is the WMMA above
            acc[nn][0] += a.x * lds_w[(nn * 16 + m) * KT + ka] +
                          a.y * lds_w[(nn * 16 + m) * KT + ka + 1];
        }
#endif
    }

    #pragma unroll
    for (int nn = 0; nn < 4; ++nn) {
        const int n = nbase + nn * 16 + m;      // N index == lane&15 for all 8 rows
        float bb = bias ? bias[n] : 0.0f;
        #pragma unroll
        for (int r = 0; r < 8; ++r) {
            int mo = r + 8 * half;
            float vout = acc[nn][r] + bb;
            if (ACT == 1) vout = sigf(vout);
            size_t p = (size_t)(p0 + mo);
            if (!PLANAR) {
                D[p * N + n] = vout;
            } else {
                int b   = (int)(p / (HS * WS));
                int rem = (int)(p % (HS * WS));
                int i = rem / WS, j = rem % WS;
                D[(((size_t)b * N + n) * HS + i) * WS + j] = vout;
            }
        }
    }
}

// --------------------------------------------------- depthwise 3x3 + SiLU ---
__global__ void dwconv_silu_kernel(const float* __restrict__ yin,
                                   const float* __restrict__ wdw,
                                   float* __restrict__ yconv) {
    size_t idx = (size_t)blockIdx.x * blockDim.x + threadIdx.x;
    if (idx >= (size_t)NPOS * INNER) return;
    int c = (int)(idx & (INNER - 1));
    size_t p = idx >> 7;
    int b   = (int)(p / (HS * WS));
    int rem = (int)(p % (HS * WS));
    int i = rem / WS, j = rem % WS;

    float s = 0.0f;
    #pragma unroll
    for (int ky = 0; ky < 3; ++ky) {
        int ii = i + ky - 1;
        if (ii < 0 || ii >= HS) continue;
        #pragma unroll
        for (int kx = 0; kx < 3; ++kx) {
            int jj = j + kx - 1;
            if (jj < 0 || jj >= WS) continue;
            size_t pp = ((size_t)b * HS + ii) * WS + jj;
            s += yin[pp * CDIM + c] * wdw[c * 9 + ky * 3 + kx];
        }
    }
    yconv[p * INNER + c] = s * sigf(s);   // SiLU
}

// ------------------------------------------------------------- scans --------
__global__ void row_scan_kernel(const float* __restrict__ Dv, const float* __restrict__ Vv,
                                float* __restrict__ mix) {
    int idx = blockIdx.x * blockDim.x + threadIdx.x;
    if (idx >= NB * HS * INNER) return;
    int c = idx & (INNER - 1);
    int n = idx >> 7;                       // b*HS + i
    size_t base = (size_t)n * WS * INNER + c;
    float s = 0.0f;
    for (int j = 0; j < WS; ++j) {
        size_t off = base + (size_t)j * INNER;
        float d = Dv[off], v = Vv[off];
        s = (1.0f - d) * s + d * v;
        mix[off] = 0.25f * s;
    }
    s = 0.0f;
    for (int j = WS - 1; j >= 0; --j) {
        size_t off = base + (size_t)j * INNER;
        float d = Dv[off], v = Vv[off];
        s = (1.0f - d) * s + d * v;
        mix[off] += 0.25f * s;
    }
}

__global__ void col_scan_kernel(const float* __restrict__ Dv, const float* __restrict__ Vv,
                                float* __restrict__ mix) {
    int idx = blockIdx.x * blockDim.x + threadIdx.x;
    if (idx >= NB * WS * INNER) return;
    int c = idx & (INNER - 1);
    int t = idx >> 7;
    int b = t / WS, j = t % WS;
    size_t base = ((size_t)b * HS * WS + j) * INNER + c;
    const size_t stride = (size_t)WS * INNER;
    float s = 0.0f;
    for (int i = 0; i < HS; ++i) {
        size_t off = base + (size_t)i * stride;
        float d = Dv[off], v = Vv[off];
        s = (1.0f - d) * s + d * v;
        mix[off] += 0.25f * s;
    }
    s = 0.0f;
    for (int i = HS - 1; i >= 0; --i) {
        size_t off = base + (size_t)i * stride;
        float d = Dv[off], v = Vv[off];
        s = (1.0f - d) * s + d * v;
        mix[off] += 0.25f * s;
    }
}

// -------------------------------------------- bilinear x2 + residual add ----
__global__ void resize_add_kernel(const float* __restrict__ x, const float* __restrict__ small,
                                  float* __restrict__ out) {
    size_t idx = (size_t)blockIdx.x * blockDim.x + threadIdx.x;
    const size_t total = (size_t)NB * CDIM * HH * WW;
    if (idx >= total) return;
    int xx = (int)(idx % WW); size_t t = idx / WW;
    int yy = (int)(t % HH);   t /= HH;
    int cc = (int)(t % CDIM); int b = (int)(t / CDIM);

    float sy = 0.5f * yy - 0.25f;
    float sx = 0.5f * xx - 0.25f;
    int i0 = (int)floorf(sy); float fy = sy - (float)i0;
    int j0 = (int)floorf(sx); float fx = sx - (float)j0;
    int i0c = i0 < 0 ? 0 : (i0 > HS - 1 ? HS - 1 : i0);
    int i1c = (i0 + 1) < 0 ? 0 : ((i0 + 1) > HS - 1 ? HS - 1 : i0 + 1);
    int j0c = j0 < 0 ? 0 : (j0 > WS - 1 ? WS - 1 : j0);
    int j1c = (j0 + 1) < 0 ? 0 : ((j0 + 1) > WS - 1 ? WS - 1 : j0 + 1);

    const float* sp = small + ((size_t)b * CDIM + cc) * (HS * WS);
    float v00 = sp[(size_t)i0c * WS + j0c], v01 = sp[(size_t)i0c * WS + j1c];
    float v10 = sp[(size_t)i1c * WS + j0c], v11 = sp[(size_t)i1c * WS + j1c];
    float v = (1.0f - fy) * ((1.0f - fx) * v00 + fx * v01) +
              fy          * ((1.0f - fx) * v10 + fx * v11);
    out[idx] = x[idx] + v;
}

// ---------------------------------------------------------------------------
extern "C" void kernel_launch(void* const* d_in, const int* in_sizes, int n_in,
                              void* d_out, int out_size, void* d_ws, size_t ws_size,
                              hipStream_t stream) {
    const float* x      = (const float*)d_in[0];
    const float* gn_w   = (const float*)d_in[1];
    const float* gn_b   = (const float*)d_in[2];
    const float* w_in   = (const float*)d_in[3];
    const float* w_dw   = (const float*)d_in[4];
    const float* row_wd = (const float*)d_in[5];
    const float* row_bd = (const float*)d_in[6];
    const float* row_wv = (const float*)d_in[7];
    const float* row_bv = (const float*)d_in[8];
    const float* col_wd = (const float*)d_in[9];
    const float* col_bd = (const float*)d_in[10];
    const float* col_wv = (const float*)d_in[11];
    const float* col_bv = (const float*)d_in[12];
    const float* w_out  = (const float*)d_in[13];
    float* out = (float*)d_out;

    float* ws     = (float*)d_ws;
    float* stats  = ws;                             // 16 floats
    float* pooled = ws + 16;                        // NPOS*256
    float* yin    = pooled + (size_t)NPOS * CDIM;   // NPOS*256
    float* yconv  = yin    + (size_t)NPOS * CDIM;   // NPOS*128
    float* Dbuf   = yconv  + (size_t)NPOS * INNER;  // NPOS*128
    float* Vbuf   = Dbuf   + (size_t)NPOS * INNER;  // NPOS*128
    float* mixb   = Vbuf   + (size_t)NPOS * INNER;  // NPOS*128
    float* out_small = pooled;                      // reuse after w_in GEMM

    zero_stats_kernel<<<1, 32, 0, stream>>>(stats);
    reduce_stats_kernel<<<NB * 160, 256, 0, stream>>>(x, stats);
    pool_norm_kernel<<<(NB * CDIM * HS * WS) / 256, 256, 0, stream>>>(x, stats, gn_w, gn_b, pooled);

    // w_in: (p,256)x(256->256) -> yin; [0,128)=conv input, [128,256)=gate
    gemm_wmma_kernel<256, false, 0, false><<<dim3(NPOS / 64, 4), 128, 0, stream>>>(
        pooled, w_in, nullptr, nullptr, yin, CDIM);

    dwconv_silu_kernel<<<(NPOS * INNER) / 256, 256, 0, stream>>>(yin, w_dw, yconv);

    // row projections + scans
    gemm_wmma_kernel<128, false, 1, false><<<dim3(NPOS / 64, 2), 128, 0, stream>>>(
        yconv, row_wd, row_bd, nullptr, Dbuf, INNER);
    gemm_wmma_kernel<128, false, 0, false><<<dim3(NPOS / 64, 2), 128, 0, stream>>>(
        yconv, row_wv, row_bv, nullptr, Vbuf, INNER);
    row_scan_kernel<<<(NB * HS * INNER) / 256, 256, 0, stream>>>(Dbuf, Vbuf, mixb);

    // col projections + scans (reuse D/V buffers)
    gemm_wmma_kernel<128, false, 1, false><<<dim3(NPOS / 64, 2), 128, 0, stream>>>(
        yconv, col_wd, col_bd, nullptr, Dbuf, INNER);
    gemm_wmma_kernel<128, false, 0, false><<<dim3(NPOS / 64, 2), 128, 0, stream>>>(
        yconv, col_wv, col_bv, nullptr, Vbuf, INNER);
    col_scan_kernel<<<(NB * WS * INNER) / 256, 256, 0, stream>>>(Dbuf, Vbuf, mixb);

    // gated w_out GEMM (gate fused into A-load), planar store
    gemm_wmma_kernel<128, true, 0, true><<<dim3(NPOS / 64, 4), 128, 0, stream>>>(
        mixb, w_out, nullptr, yin + INNER, out_small, CDIM);

    resize_add_kernel<<<(int)(((size_t)NB * CDIM * HH * WW + 255) / 256), 256, 0, stream>>>(
        x, out_small, out);
}